// ProxyClassifyNet_79731772883628
// MI455X (gfx1250) — compile-verified
//
#include <hip/hip_runtime.h>
#include <hip/hip_bf16.h>

typedef __attribute__((ext_vector_type(2))) float v2f;
typedef __attribute__((ext_vector_type(8))) float v8f;

#define B_SZ   256
#define INDIM  1024
#define DDIM   512
#define PDIM   2048

// ---------------------------------------------------------------------------
// feat = x @ Wb^T + bb   (M=256, N=512, K=1024), plus per-row sum of squares.
// One wave computes a 32(M) x 64(N) slab: 2 A frags x 4 B frags -> 8 WMMAs
// per K-step, so each B fragment load feeds 2 WMMAs (B traffic halved).
// WMMA f32 16x16x4: A lanes 0-15 hold M=0..15 K={0,1}, lanes 16-31 K={2,3};
// B mirrors it with N on lanes. C/D: VGPR i -> M=i (lanes<16) / M=i+8.
// ---------------------------------------------------------------------------
__global__ void __launch_bounds__(128) gemm_feat_kernel(
    const float* __restrict__ x, const float* __restrict__ Wb,
    const float* __restrict__ bb, float* __restrict__ feat,
    float* __restrict__ fsumsq)
{
    const int wid   = blockIdx.x * 4 + (threadIdx.x >> 5);   // 64 waves total
    const int lane  = threadIdx.x & 31;
    const int half  = lane >> 4;
    const int lid   = lane & 15;
    const int mBase = (wid >> 3) * 32;        // 8 m-supertiles of 32 rows
    const int nBase = (wid & 7) * 64;         // 8 n-groups of 64 cols

    const float* a0Ptr = x  + (size_t)(mBase + lid) * INDIM + 2 * half;
    const float* a1Ptr = a0Ptr + (size_t)16 * INDIM;
    const float* bPtr  = Wb + (size_t)(nBase + lid) * INDIM + 2 * half;

    const v8f z = {0.f,0.f,0.f,0.f,0.f,0.f,0.f,0.f};
    v8f c[2][4] = {{z, z, z, z}, {z, z, z, z}};

    for (int k = 0; k < INDIM; k += 4) {
        __builtin_prefetch(a0Ptr + k + 256, 0, 1);
        __builtin_prefetch(bPtr  + k + 256, 0, 1);
        v2f a0 = *(const v2f*)(a0Ptr + k);
        v2f a1 = *(const v2f*)(a1Ptr + k);
#pragma unroll
        for (int t = 0; t < 4; ++t) {
            v2f b = *(const v2f*)(bPtr + k + t * 16 * INDIM);
            c[0][t] = __builtin_amdgcn_wmma_f32_16x16x4_f32(
                false, a0, false, b, (short)0, c[0][t], false, false);
            c[1][t] = __builtin_amdgcn_wmma_f32_16x16x4_f32(
                false, a1, false, b, (short)0, c[1][t], false, false);
        }
    }

    float rowsq[2][8];
#pragma unroll
    for (int mt = 0; mt < 2; ++mt)
#pragma unroll
        for (int i = 0; i < 8; ++i) rowsq[mt][i] = 0.f;

#pragma unroll
    for (int t = 0; t < 4; ++t) {
        const int n = nBase + 16 * t + lid;
        const float bias = bb[n];
#pragma unroll
        for (int mt = 0; mt < 2; ++mt) {
#pragma unroll
            for (int i = 0; i < 8; ++i) {
                const int m = mBase + 16 * mt + i + 8 * half;
                const float v = c[mt][t][i] + bias;
                feat[(size_t)m * DDIM + n] = v;
                rowsq[mt][i] += v * v;
            }
        }
    }
    // reduce across the 16 lanes that share a row (xor stays within half-wave)
#pragma unroll
    for (int mt = 0; mt < 2; ++mt) {
#pragma unroll
        for (int i = 0; i < 8; ++i) {
            float s = rowsq[mt][i];
            s += __shfl_xor(s, 1, 32);
            s += __shfl_xor(s, 2, 32);
            s += __shfl_xor(s, 4, 32);
            s += __shfl_xor(s, 8, 32);
            if (lid == 0)
                atomicAdd(&fsumsq[mBase + 16 * mt + i + 8 * half], s);
        }
    }
}

// ---------------------------------------------------------------------------
// Fused dual GEMM on feat (M=256, K=512, N=2048):
//   out = feat @ Wm^T + bm                         -> d_out
//   den = fsumsq + pnorm - 2 * feat @ proxies^T    -> workspace
// One wave: 32(M) x 64(N); 2 A frags feed 16 WMMAs (8 Wm + 8 proxies)
// per K-step against 4 Wm frags + 4 proxies frags.
// ---------------------------------------------------------------------------
__global__ void __launch_bounds__(128) gemm_heads_kernel(
    const float* __restrict__ feat, const float* __restrict__ Wm,
    const float* __restrict__ bm,   const float* __restrict__ proxies,
    const float* __restrict__ pnorm,const float* __restrict__ fsumsq,
    float* __restrict__ out, float* __restrict__ den)
{
    const int wid   = blockIdx.x * 4 + (threadIdx.x >> 5);   // 256 waves total
    const int lane  = threadIdx.x & 31;
    const int half  = lane >> 4;
    const int lid   = lane & 15;
    const int mBase = (wid >> 5) * 32;        // 8 m-supertiles of 32 rows
    const int nBase = (wid & 31) * 64;        // 32 n-groups of 64 cols

    const float* a0Ptr = feat + (size_t)(mBase + lid) * DDIM + 2 * half;
    const float* a1Ptr = a0Ptr + (size_t)16 * DDIM;
    const float* wPtr  = Wm      + (size_t)(nBase + lid) * DDIM + 2 * half;
    const float* pPtr  = proxies + (size_t)(nBase + lid) * DDIM + 2 * half;

    const v8f z = {0.f,0.f,0.f,0.f,0.f,0.f,0.f,0.f};
    v8f cW[2][4] = {{z, z, z, z}, {z, z, z, z}};
    v8f cP[2][4] = {{z, z, z, z}, {z, z, z, z}};

    for (int k = 0; k < DDIM; k += 4) {
        __builtin_prefetch(a0Ptr + k + 256, 0, 1);
        __builtin_prefetch(wPtr  + k + 256, 0, 1);
        __builtin_prefetch(pPtr  + k + 256, 0, 1);
        v2f a0 = *(const v2f*)(a0Ptr + k);
        v2f a1 = *(const v2f*)(a1Ptr + k);
#pragma unroll
        for (int t = 0; t < 4; ++t) {
            v2f bw = *(const v2f*)(wPtr + k + t * 16 * DDIM);
            v2f bp = *(const v2f*)(pPtr + k + t * 16 * DDIM);
            cW[0][t] = __builtin_amdgcn_wmma_f32_16x16x4_f32(
                false, a0, false, bw, (short)0, cW[0][t], false, false);
            cW[1][t] = __builtin_amdgcn_wmma_f32_16x16x4_f32(
                false, a1, false, bw, (short)0, cW[1][t], false, false);
            cP[0][t] = __builtin_amdgcn_wmma_f32_16x16x4_f32(
                false, a0, false, bp, (short)0, cP[0][t], false, false);
            cP[1][t] = __builtin_amdgcn_wmma_f32_16x16x4_f32(
                false, a1, false, bp, (short)0, cP[1][t], false, false);
        }
    }

    float fs[2][8];
#pragma unroll
    for (int mt = 0; mt < 2; ++mt)
#pragma unroll
        for (int i = 0; i < 8; ++i)
            fs[mt][i] = fsumsq[mBase + 16 * mt + i + 8 * half];

#pragma unroll
    for (int t = 0; t < 4; ++t) {
        const int n = nBase + 16 * t + lid;
        const float bmv = bm[n];
        const float pnv = pnorm[n];
#pragma unroll
        for (int mt = 0; mt < 2; ++mt) {
#pragma unroll
            for (int i = 0; i < 8; ++i) {
                const int m = mBase + 16 * mt + i + 8 * half;
                out[(size_t)m * PDIM + n] = cW[mt][t][i] + bmv;
                den[(size_t)m * PDIM + n] = fs[mt][i] + pnv - 2.0f * cP[mt][t][i];
            }
        }
    }
}

// ---------------------------------------------------------------------------
__global__ void __launch_bounds__(256) pnorm_kernel(
    const float* __restrict__ proxies, float* __restrict__ pnorm)
{
    const int p = blockIdx.x * 256 + threadIdx.x;      // 2048 threads
    const float4* row = (const float4*)(proxies + (size_t)p * DDIM);
    float s = 0.f;
    for (int i = 0; i < DDIM / 4; ++i) {
        float4 v = row[i];
        s += v.x * v.x + v.y * v.y + v.z * v.z + v.w * v.w;
    }
    pnorm[p] = s;
}

__global__ void zero_fsumsq_kernel(float* __restrict__ fsumsq)
{
    fsumsq[threadIdx.x] = 0.f;
}

// ---------------------------------------------------------------------------
// Per-row: rowloss[b] = den[b, y_b] + logsumexp_p(-den[b, p])
//                     = den[b, y_b] - min_p(den) + log(sum exp(min - den))
// ---------------------------------------------------------------------------
__global__ void __launch_bounds__(256) loss_row_kernel(
    const float* __restrict__ den, const int* __restrict__ y,
    float* __restrict__ rowloss)
{
    const int b = blockIdx.x;
    const int t = threadIdx.x;
    const float* row = den + (size_t)b * PDIM;
    __shared__ float red[256];

    float mn = 3.402823466e+38f;
    for (int p = t; p < PDIM; p += 256) mn = fminf(mn, row[p]);
    red[t] = mn; __syncthreads();
    for (int s = 128; s > 0; s >>= 1) {
        if (t < s) red[t] = fminf(red[t], red[t + s]);
        __syncthreads();
    }
    mn = red[0];
    __syncthreads();

    float sum = 0.f;
    for (int p = t; p < PDIM; p += 256) sum += expf(mn - row[p]);
    red[t] = sum; __syncthreads();
    for (int s = 128; s > 0; s >>= 1) {
        if (t < s) red[t] += red[t + s];
        __syncthreads();
    }
    if (t == 0) rowloss[b] = row[y[b]] - mn + logf(red[0]);
}

__global__ void __launch_bounds__(256) finalize_kernel(
    const float* __restrict__ rowloss, const float* __restrict__ fsumsq,
    float* __restrict__ scalars)   // scalars[0]=loss, scalars[1]=reg_e
{
    const int t = threadIdx.x;
    __shared__ float r1[256], r2[256];
    r1[t] = rowloss[t];
    r2[t] = sqrtf(fsumsq[t]);
    __syncthreads();
    for (int s = 128; s > 0; s >>= 1) {
        if (t < s) { r1[t] += r1[t + s]; r2[t] += r2[t + s]; }
        __syncthreads();
    }
    if (t == 0) {
        scalars[0] = r1[0] * (1.0f / B_SZ);
        scalars[1] = r2[0] * (1.0f / B_SZ);
    }
}

// ---------------------------------------------------------------------------
extern "C" void kernel_launch(void* const* d_in, const int* in_sizes, int n_in,
                              void* d_out, int out_size, void* d_ws, size_t ws_size,
                              hipStream_t stream)
{
    const float* x       = (const float*)d_in[0];
    const int*   y       = (const int*)  d_in[1];
    const float* Wb      = (const float*)d_in[2];
    const float* bb      = (const float*)d_in[3];
    const float* Wm      = (const float*)d_in[4];
    const float* bm      = (const float*)d_in[5];
    const float* proxies = (const float*)d_in[6];

    float* out = (float*)d_out;                       // [256*2048] then loss, reg_e
    char*  ws  = (char*)d_ws;
    float* feat    = (float*)(ws);                    // 256*512   = 512 KB
    float* fsumsq  = (float*)(ws + 524288);           // 256 f32
    float* pnorm   = (float*)(ws + 525312);           // 2048 f32
    float* den     = (float*)(ws + 533504);           // 256*2048  = 2 MB
    float* rowloss = (float*)(ws + 2630656);          // 256 f32

    zero_fsumsq_kernel<<<1, 256, 0, stream>>>(fsumsq);
    pnorm_kernel<<<PDIM / 256, 256, 0, stream>>>(proxies, pnorm);
    gemm_feat_kernel<<<16, 128, 0, stream>>>(x, Wb, bb, feat, fsumsq);
    gemm_heads_kernel<<<64, 128, 0, stream>>>(feat, Wm, bm, proxies, pnorm,
                                              fsumsq, out, den);
    loss_row_kernel<<<B_SZ, 256, 0, stream>>>(den, y, rowloss);
    finalize_kernel<<<1, 256, 0, stream>>>(rowloss, fsumsq, out + (size_t)B_SZ * PDIM);
}